// SS2Depth_K1_45243185496320
// MI455X (gfx1250) — compile-verified
//
#include <hip/hip_runtime.h>
#include <hip/hip_bf16.h>

// ---------------------------------------------------------------------------
// SS2D (VMamba) block for MI455X / gfx1250, fp32.
//  - GEMMs via V_WMMA_F32_16X16X4_F32 (full fp32 precision; compute is not
//    the bottleneck at ~2 GFLOP / ~35MB L2-resident working set).
//  - Selective scan double-buffers its 38x32 x_dbl tile in LDS via the
//    Tensor Data Mover (tensor_load_to_lds + s_wait_tensorcnt), overlapping
//    the next chunk's fill with the current chunk's exp/fma chain.
// ---------------------------------------------------------------------------

#define BATCH     2
#define HH        64
#define WW        64
#define DM        96
#define DI        192
#define NS        16
#define RK        6
#define LSP       4096           // spatial tokens (64*64)
#define LTOT      4288           // 192 depth tokens + 4096 spatial
#define L2SP      1024           // 32*32 after stride-2 conv

typedef __attribute__((ext_vector_type(2))) float        v2f;
typedef __attribute__((ext_vector_type(8))) float        v8f;
typedef __attribute__((ext_vector_type(4))) unsigned int u32x4;
typedef __attribute__((ext_vector_type(4))) int          i32x4;
typedef __attribute__((ext_vector_type(8))) int          i32x8;

__device__ __forceinline__ float siluf(float v) {
    return v / (1.0f + __expf(-v));
}

// ---------------------------------------------------------------------------
// WMMA tile helper (see round-1; codegen-confirmed v_wmma_f32_16x16x4_f32)
// ---------------------------------------------------------------------------
#if __has_builtin(__builtin_amdgcn_wmma_f32_16x16x4_f32)
#define HAVE_WMMA_F32X4 1
#endif

__device__ __forceinline__ v8f wmma_tile(const float* __restrict__ aBase,
                                         int rsA, int csA,
                                         const float* __restrict__ wBase,
                                         int ldw, int n0, int validN, int K)
{
    const int lane = threadIdx.x & 31;
    const int lm   = lane & 15;
    const int lh   = lane >> 4;
    v8f acc = {0.f, 0.f, 0.f, 0.f, 0.f, 0.f, 0.f, 0.f};
#ifdef HAVE_WMMA_F32X4
    const bool nOK = (n0 + lm) < validN;
    const float* wRow = wBase + (long)(n0 + lm) * ldw;
    for (int k = 0; k < K; k += 4) {
        const int ka = k + 2 * lh;
        v2f a, b;
        a[0] = aBase[(long)lm * rsA + (long)ka * csA];
        a[1] = aBase[(long)lm * rsA + (long)(ka + 1) * csA];
        b[0] = nOK ? wRow[ka]     : 0.f;
        b[1] = nOK ? wRow[ka + 1] : 0.f;
        acc = __builtin_amdgcn_wmma_f32_16x16x4_f32(
                  false, a, false, b, (short)0, acc, false, false);
    }
#else
    const int n = n0 + lm;
    if (n < validN) {
        for (int v = 0; v < 8; ++v) {
            const int m = v + 8 * lh;
            float s = 0.f;
            for (int k = 0; k < K; ++k)
                s += aBase[(long)m * rsA + (long)k * csA] * wBase[(long)n * ldw + k];
            acc[v] = s;
        }
    }
#endif
    return acc;
}

// ---------------------------------------------------------------------------
// Tensor Data Mover staging of one 38x32 fp32 tile of x_dbl into LDS.
//   tensor: 38 rows x 4288 cols row-major, tile starts at column l0.
//   D# group0: count=1 | lds_addr | global_addr | type=2
//   D# group1: data_size=4B, dim0=4288, dim1=38, tile0=32, tile1=38,
//              dim0_stride=4288  (groups 2/3 zero: 2-D tensor)
// ---------------------------------------------------------------------------
#if __has_builtin(__builtin_amdgcn_tensor_load_to_lds)
#define HAVE_TDM 1
#endif
#if defined(__has_include)
#if __has_include(<hip/amd_detail/amd_gfx1250_TDM.h>)
#define TDM_6ARG 1            // therock headers => 6-arg builtin
#endif
#endif

__device__ __forceinline__ void stage_tile(const float* __restrict__ xd,
                                           int l0, float* dst, int tid)
{
#ifdef HAVE_TDM
    if (tid == 0) {
        const unsigned long long ga = (unsigned long long)(xd + l0);
        const unsigned ldsaddr = (unsigned)(unsigned long long)dst;
        u32x4 g0 = { 1u,                                   // count=1
                     ldsaddr,                              // lds_addr
                     (unsigned)(ga & 0xFFFFFFFFu),         // global_addr lo
                     (unsigned)((ga >> 32) & 0x01FFFFFFu)
                         | 0x80000000u };                  // addr hi | type=2
        i32x8 g1 = { 0x00020000,                           // data_size=4B
                     (int)((LTOT & 0xFFFF) << 16),         // tensor_dim0 lo16
                     (int)((LTOT >> 16) | (38 << 16)),     // dim0 hi | dim1 lo
                     (int)(32 << 16),                      // dim1 hi=0 | tile_dim0=32
                     38,                                   // tile_dim1=38, tile_dim2=0
                     LTOT,                                 // dim0_stride lo32
                     0, 0 };                               // stride hi, dim1_stride
        i32x4 z4 = { 0, 0, 0, 0 };
#ifdef TDM_6ARG
        i32x8 z8 = { 0, 0, 0, 0, 0, 0, 0, 0 };
        __builtin_amdgcn_tensor_load_to_lds(g0, g1, z4, z4, z8, 0);
#else
        __builtin_amdgcn_tensor_load_to_lds(g0, g1, z4, z4, 0);
#endif
    }
#else
    // Fallback: per-lane async LDS fill (ASYNCcnt path).
    for (int idx = tid; idx < 38 * 8; idx += DI) {
        const int r = idx >> 3;
        const int i = (idx & 7) * 4;
        const float* ga = xd + (long)r * LTOT + l0 + i;
        const unsigned lo = (unsigned)(unsigned long long)(dst + r * 32 + i);
        asm volatile("global_load_async_to_lds_b128 %0, %1, off"
                     :: "v"(lo), "v"(ga) : "memory");
    }
#endif
}

__device__ __forceinline__ void stage_wait(int tid)
{
#ifdef HAVE_TDM
    if (tid == 0) __builtin_amdgcn_s_wait_tensorcnt((short)0);
#else
#if __has_builtin(__builtin_amdgcn_s_wait_asynccnt)
    __builtin_amdgcn_s_wait_asynccnt(0);
#else
    asm volatile("s_wait_asynccnt 0" ::: "memory");
#endif
#endif
}

// ---------------------------------------------------------------------------
// K1: xz = x @ in_proj_w^T     (M=8192 pixels, N=384, K=96)
// ---------------------------------------------------------------------------
__global__ void k_inproj(const float* __restrict__ x,
                         const float* __restrict__ w,
                         float* __restrict__ x1c,
                         float* __restrict__ zbuf)
{
    const int wv = blockIdx.x * 4 + (threadIdx.x >> 5);
    if (wv >= 512 * 24) return;
    const int m0 = (wv % 512) * 16;
    const int n0 = (wv / 512) * 16;
    v8f acc = wmma_tile(x + (long)m0 * DM, DM, 1, w, DM, n0, 2 * DI, DM);
    const int lane = threadIdx.x & 31, lm = lane & 15, lh = lane >> 4;
    const int n = n0 + lm;
#pragma unroll
    for (int v = 0; v < 8; ++v) {
        const int m  = m0 + v + 8 * lh;
        const int b  = m >> 12;
        const int hw = m & (LSP - 1);
        const float val = acc[v];
        if (n < DI) x1c[((long)(b * DI + n)) * LSP + hw] = val;
        else        zbuf[(long)m * DI + (n - DI)] = val;
    }
}

// ---------------------------------------------------------------------------
// K2: depthwise 3x3 (pad 1) + bias + SiLU; writes xc into xs[:, :, 192+hw]
// ---------------------------------------------------------------------------
__global__ void k_conv(const float* __restrict__ x1c,
                       const float* __restrict__ cw,
                       const float* __restrict__ cb,
                       float* __restrict__ xs)
{
    const long i = (long)blockIdx.x * 256 + threadIdx.x;
    if (i >= (long)BATCH * DI * LSP) return;
    const int hw = (int)(i & (LSP - 1));
    const int bc = (int)(i >> 12);
    const int c  = bc % DI;
    const int y  = hw >> 6, xo = hw & 63;
    const float* plane = x1c + (long)bc * LSP;
    float acc = cb[c];
#pragma unroll
    for (int ky = 0; ky < 3; ++ky) {
        const int iy = y + ky - 1;
        if (iy < 0 || iy >= HH) continue;
#pragma unroll
        for (int kx = 0; kx < 3; ++kx) {
            const int ix = xo + kx - 1;
            if (ix < 0 || ix >= WW) continue;
            acc += plane[iy * WW + ix] * cw[c * 9 + ky * 3 + kx];
        }
    }
    xs[(long)bc * LTOT + DI + hw] = siluf(acc);
}

// ---------------------------------------------------------------------------
// K3: depthwise 3x3 stride-2 (pad 1) + bias -> dxc channel-major (B,192,1024)
// ---------------------------------------------------------------------------
__global__ void k_dconv(const float* __restrict__ xs,
                        const float* __restrict__ dw,
                        const float* __restrict__ db,
                        float* __restrict__ dxc)
{
    const long i = (long)blockIdx.x * 256 + threadIdx.x;
    if (i >= (long)BATCH * DI * L2SP) return;
    const int o  = (int)(i & (L2SP - 1));
    const int bc = (int)(i >> 10);
    const int c  = bc % DI;
    const int oy = o >> 5, ox = o & 31;
    const float* plane = xs + (long)bc * LTOT + DI;
    float acc = db[c];
#pragma unroll
    for (int ky = 0; ky < 3; ++ky) {
        const int iy = 2 * oy + ky - 1;
        if (iy < 0 || iy >= HH) continue;
#pragma unroll
        for (int kx = 0; kx < 3; ++kx) {
            const int ix = 2 * ox + kx - 1;
            if (ix < 0 || ix >= WW) continue;
            acc += plane[iy * WW + ix] * dw[c * 9 + ky * 3 + kx];
        }
    }
    dxc[(long)bc * L2SP + o] = acc;
}

// ---------------------------------------------------------------------------
// K4: depth_fc (M=192, N=192, K=1024); transpose+bias+SiLU fused into store
// ---------------------------------------------------------------------------
__global__ void k_depthfc(const float* __restrict__ dxc,
                          const float* __restrict__ w,
                          const float* __restrict__ bias,
                          float* __restrict__ xs)
{
    const int wv = blockIdx.x * 4 + (threadIdx.x >> 5);
    if (wv >= BATCH * 12 * 12) return;
    const int b  = wv / 144;
    const int t  = wv % 144;
    const int m0 = (t % 12) * 16;
    const int n0 = (t / 12) * 16;
    v8f acc = wmma_tile(dxc + (long)b * DI * L2SP + (long)m0 * L2SP,
                        L2SP, 1, w, L2SP, n0, DI, L2SP);
    const int lane = threadIdx.x & 31, lm = lane & 15, lh = lane >> 4;
    const int n = n0 + lm;
    const float bb = bias[n];
#pragma unroll
    for (int v = 0; v < 8; ++v) {
        const int m = m0 + v + 8 * lh;
        xs[((long)(b * DI + n)) * LTOT + m] = siluf(acc[v] + bb);
    }
}

// ---------------------------------------------------------------------------
// K5: x_dbl[b,c,l] = sum_d xs[b,d,l] * x_proj_w[c,d]  (M=4288, N=38, K=192)
// ---------------------------------------------------------------------------
__global__ void k_xproj(const float* __restrict__ xs,
                        const float* __restrict__ w,
                        float* __restrict__ xdbl)
{
    const int wv = blockIdx.x * 4 + (threadIdx.x >> 5);
    if (wv >= BATCH * 268 * 3) return;
    const int b  = wv / (268 * 3);
    const int t  = wv % (268 * 3);
    const int m0 = (t % 268) * 16;
    const int n0 = (t / 268) * 16;
    v8f acc = wmma_tile(xs + (long)b * DI * LTOT + m0, 1, LTOT,
                        w, DI, n0, RK + 2 * NS, DI);
    const int lane = threadIdx.x & 31, lm = lane & 15, lh = lane >> 4;
    const int n = n0 + lm;
    if (n >= RK + 2 * NS) return;
#pragma unroll
    for (int v = 0; v < 8; ++v) {
        const int m = m0 + v + 8 * lh;
        xdbl[((long)(b * (RK + 2 * NS) + n)) * LTOT + m] = acc[v];
    }
}

// ---------------------------------------------------------------------------
// K6: selective scan, TDM double-buffered LDS staging.
// ---------------------------------------------------------------------------
__global__ __launch_bounds__(DI) void k_scan(const float* __restrict__ xs,
                                             const float* __restrict__ xdbl,
                                             const float* __restrict__ dtw,
                                             const float* __restrict__ dtb,
                                             const float* __restrict__ Alogs,
                                             const float* __restrict__ Ds,
                                             float* __restrict__ ybuf)
{
    const int b = blockIdx.x;
    const int d = threadIdx.x;
    __shared__ float sx[2][38 * 32];

    float h[NS], Ar[NS], wr[RK];
#pragma unroll
    for (int n = 0; n < NS; ++n) { h[n] = 0.f; Ar[n] = -__expf(Alogs[d * NS + n]); }
#pragma unroll
    for (int r = 0; r < RK; ++r) wr[r] = dtw[d * RK + r];
    const float dbias = dtb[d];
    const float Dd    = Ds[d];

    const float* urow = xs   + ((long)b * DI + d) * LTOT;
    const float* xd   = xdbl + (long)b * (RK + 2 * NS) * LTOT;
    float*       yb   = ybuf + (long)b * LSP * DI;

    stage_tile(xd, 0, sx[0], d);          // prologue fill of buffer 0
    stage_wait(d);
    __syncthreads();

    for (int c = 0; c < LTOT / 32; ++c) {
        const int l0 = c * 32;
        const float* cur = sx[c & 1];
        if (l0 + 32 < LTOT)               // overlap next fill with compute
            stage_tile(xd, l0 + 32, sx[(c + 1) & 1], d);
        __builtin_prefetch(urow + l0 + 32, 0, 0);   // global_prefetch_b8

        for (int i = 0; i < 32; ++i) {
            const int l = l0 + i;
            const float u = urow[l];
            float draw = dbias;
#pragma unroll
            for (int r = 0; r < RK; ++r) draw += wr[r] * cur[r * 32 + i];
            const float delta = (draw > 20.f) ? draw : log1pf(__expf(draw));
            const float du = delta * u;
            float y = 0.f;
#pragma unroll
            for (int n = 0; n < NS; ++n) {
                const float dA = __expf(delta * Ar[n]);
                h[n] = dA * h[n] + du * cur[(RK + n) * 32 + i];
                y   += h[n] * cur[(RK + NS + n) * 32 + i];
            }
            if (l >= DI) yb[(long)(l - DI) * DI + d] = y + Dd * u;
        }
        stage_wait(d);
        __syncthreads();
    }
}

// ---------------------------------------------------------------------------
// K7: LayerNorm over DI per pixel + gate with silu(z).
// ---------------------------------------------------------------------------
__global__ __launch_bounds__(DI) void k_lngate(const float* __restrict__ ybuf,
                                               const float* __restrict__ zbuf,
                                               const float* __restrict__ nw,
                                               const float* __restrict__ nb,
                                               float* __restrict__ g)
{
    const int p = blockIdx.x;
    const int d = threadIdx.x;
    __shared__ float s1[256], s2[256];
    const float val = ybuf[(long)p * DI + d];
    s1[d] = val;
    s2[d] = val * val;
    if (d < 256 - DI) { s1[DI + d] = 0.f; s2[DI + d] = 0.f; }
    __syncthreads();
    for (int s = 128; s > 0; s >>= 1) {
        if (d < s) { s1[d] += s1[d + s]; s2[d] += s2[d + s]; }
        __syncthreads();
    }
    const float mu  = s1[0] * (1.0f / DI);
    const float var = s2[0] * (1.0f / DI) - mu * mu;
    const float r   = rsqrtf(var + 1e-5f);
    const float z   = zbuf[(long)p * DI + d];
    g[(long)p * DI + d] = ((val - mu) * r * nw[d] + nb[d]) * siluf(z);
}

// ---------------------------------------------------------------------------
// K8: out = g @ out_proj_w^T   (M=8192, N=96, K=192)
// ---------------------------------------------------------------------------
__global__ void k_outproj(const float* __restrict__ g,
                          const float* __restrict__ w,
                          float* __restrict__ out)
{
    const int wv = blockIdx.x * 4 + (threadIdx.x >> 5);
    if (wv >= 512 * 6) return;
    const int m0 = (wv % 512) * 16;
    const int n0 = (wv / 512) * 16;
    v8f acc = wmma_tile(g + (long)m0 * DI, DI, 1, w, DI, n0, DM, DI);
    const int lane = threadIdx.x & 31, lm = lane & 15, lh = lane >> 4;
    const int n = n0 + lm;
#pragma unroll
    for (int v = 0; v < 8; ++v) {
        const int m = m0 + v + 8 * lh;
        out[(long)m * DM + n] = acc[v];
    }
}

// ---------------------------------------------------------------------------
extern "C" void kernel_launch(void* const* d_in, const int* in_sizes, int n_in,
                              void* d_out, int out_size, void* d_ws, size_t ws_size,
                              hipStream_t stream)
{
    const float* x        = (const float*)d_in[0];
    const float* inproj_w = (const float*)d_in[1];
    const float* conv_w   = (const float*)d_in[2];
    const float* conv_b   = (const float*)d_in[3];
    const float* dconv_w  = (const float*)d_in[4];
    const float* dconv_b  = (const float*)d_in[5];
    const float* dfc_w    = (const float*)d_in[6];
    const float* dfc_b    = (const float*)d_in[7];
    const float* xproj_w  = (const float*)d_in[8];
    const float* dt_w     = (const float*)d_in[9];
    const float* dt_b     = (const float*)d_in[10];
    const float* A_logs   = (const float*)d_in[11];
    const float* Ds       = (const float*)d_in[12];
    const float* norm_w   = (const float*)d_in[13];
    const float* norm_b   = (const float*)d_in[14];
    const float* oproj_w  = (const float*)d_in[15];
    float* out = (float*)d_out;

    float* ws   = (float*)d_ws;
    float* x1c  = ws;                                         // B*192*4096
    float* zbuf = x1c  + (long)BATCH * DI * LSP;              // B*4096*192
    float* xsb  = zbuf + (long)BATCH * LSP * DI;              // B*192*4288
    float* dxc  = xsb  + (long)BATCH * DI * LTOT;             // B*192*1024
    float* xdbl = dxc  + (long)BATCH * DI * L2SP;             // B*38*4288
    float* ybuf = xdbl + (long)BATCH * (RK + 2 * NS) * LTOT;  // B*4096*192
    float* gbuf = ybuf + (long)BATCH * LSP * DI;              // B*4096*192

    k_inproj <<<3072, 128, 0, stream>>>(x, inproj_w, x1c, zbuf);
    k_conv   <<<6144, 256, 0, stream>>>(x1c, conv_w, conv_b, xsb);
    k_dconv  <<<1536, 256, 0, stream>>>(xsb, dconv_w, dconv_b, dxc);
    k_depthfc<<<  72, 128, 0, stream>>>(dxc, dfc_w, dfc_b, xsb);
    k_xproj  <<< 402, 128, 0, stream>>>(xsb, xproj_w, xdbl);
    k_scan   <<<BATCH, DI, 0, stream>>>(xsb, xdbl, dt_w, dt_b, A_logs, Ds, ybuf);
    k_lngate <<<BATCH * LSP, DI, 0, stream>>>(ybuf, zbuf, norm_w, norm_b, gbuf);
    k_outproj<<< 768, 128, 0, stream>>>(gbuf, oproj_w, out);
}